// MatrixPredictor_4664334483959
// MI455X (gfx1250) — compile-verified
//
#include <hip/hip_runtime.h>
#include <math.h>

typedef __attribute__((ext_vector_type(2))) float v2f;
typedef __attribute__((ext_vector_type(8))) float v8f;

// D = A(16x4) * B(4x16) + C(16x16), all fp32, wave32
#define WMMA_F32_16x16x4(a, b, c) \
  __builtin_amdgcn_wmma_f32_16x16x4_f32(false, (a), false, (b), (short)0, (c), false, false)

constexpr int B_   = 16384;
constexpr int IN_  = 128;
constexpr int H_   = 512;
constexpr int Y_   = 32;
constexpr int K_   = 8;
constexpr int DVW  = Y_ + Y_ * K_;   // 288 columns: [D(32) | V(256)]

// ---------------------------------------------------------------------------
// Kernel 1: feat = relu(x @ W1 + b1)   [16384,128]x[128,512] -> [16384,512]
// block = 128 threads = 4 waves; block tile 64x64; wave tile 16x64
// ---------------------------------------------------------------------------
__global__ __launch_bounds__(128) void k_feat(const float* __restrict__ x,
                                              const float* __restrict__ W1,
                                              const float* __restrict__ b1,
                                              float* __restrict__ feat) {
  __shared__ __align__(16) float As[64 * 66];   // 64 rows x 64 K, stride 66
  __shared__ __align__(16) float Bs[64 * 66];   // 64 K x 64 cols, stride 66
  const int tid  = threadIdx.x;
  const int wave = tid >> 5;
  const int lane = tid & 31;
  const int rowBase = blockIdx.x * 64;
  const int colBase = blockIdx.y * 64;
  const int n   = lane & 15;            // A row within tile == B/D col
  const int kh  = (lane >> 4) << 1;     // 0 or 2 (K pair within WMMA step)
  const int mh  = (lane >> 4) << 3;     // 0 or 8 (D row offset)
  const int wrow = wave * 16;

  v8f acc[4] = {};

  for (int kc = 0; kc < 2; ++kc) {
    #pragma unroll
    for (int i = 0; i < 32; ++i) {              // 64*64 / 128 threads
      int idx = tid + i * 128;
      int r = idx >> 6, c = idx & 63;
      As[r * 66 + c] = x[(size_t)(rowBase + r) * IN_ + kc * 64 + c];
      Bs[r * 66 + c] = W1[(size_t)(kc * 64 + r) * H_ + colBase + c];
    }
    __syncthreads();
    #pragma unroll
    for (int s = 0; s < 16; ++s) {
      int kk = s * 4 + kh;
      v2f a = *(const v2f*)&As[(wrow + n) * 66 + kk];   // even stride+even kk -> 8B aligned
      #pragma unroll
      for (int j = 0; j < 4; ++j) {
        v2f b;
        b.x = Bs[kk * 66 + j * 16 + n];
        b.y = Bs[(kk + 1) * 66 + j * 16 + n];
        acc[j] = WMMA_F32_16x16x4(a, b, acc[j]);
      }
    }
    __syncthreads();
  }

  #pragma unroll
  for (int j = 0; j < 4; ++j) {
    int col = colBase + j * 16 + n;
    float bias = b1[col];
    #pragma unroll
    for (int v = 0; v < 8; ++v) {
      int row = rowBase + wrow + mh + v;
      float val = acc[j][v] + bias;
      feat[(size_t)row * H_ + col] = fmaxf(val, 0.0f);
    }
  }
}

// ---------------------------------------------------------------------------
// Kernel 2: DV = activation(feat @ [WD|WV] + [bD|bV])  -> [16384, 288]
//   cols [0,32):  softplus(.)+1e-3 (the D head) ; cols [32,288): raw V
// block = 128 threads = 4 waves; block tile 64 rows x 96 cols; wave 16x96
// ---------------------------------------------------------------------------
__global__ __launch_bounds__(128) void k_head(const float* __restrict__ feat,
                                              const float* __restrict__ WD,
                                              const float* __restrict__ bD,
                                              const float* __restrict__ WV,
                                              const float* __restrict__ bV,
                                              float* __restrict__ DV) {
  __shared__ __align__(16) float As[64 * 66];    // 64 rows x 64 K
  __shared__ __align__(16) float Bs[64 * 100];   // 64 K x 96 cols, stride 100
  const int tid  = threadIdx.x;
  const int wave = tid >> 5;
  const int lane = tid & 31;
  const int rowBase = blockIdx.x * 64;
  const int colBase = blockIdx.y * 96;
  const int n  = lane & 15;
  const int kh = (lane >> 4) << 1;
  const int mh = (lane >> 4) << 3;
  const int wrow = wave * 16;

  v8f acc[6] = {};

  for (int kc = 0; kc < 8; ++kc) {
    #pragma unroll
    for (int i = 0; i < 32; ++i) {
      int idx = tid + i * 128;
      int r = idx >> 6, c = idx & 63;
      As[r * 66 + c] = feat[(size_t)(rowBase + r) * H_ + kc * 64 + c];
    }
    #pragma unroll
    for (int i = 0; i < 48; ++i) {               // 64*96 / 128 threads
      int idx = tid + i * 128;
      int r = idx / 96, c = idx % 96;
      int col = colBase + c;
      int k = kc * 64 + r;
      float wv = (col < Y_) ? WD[(size_t)k * Y_ + col]
                            : WV[(size_t)k * (Y_ * K_) + (col - Y_)];
      Bs[r * 100 + c] = wv;
    }
    __syncthreads();
    #pragma unroll
    for (int s = 0; s < 16; ++s) {
      int kk = s * 4 + kh;
      v2f a = *(const v2f*)&As[(wrow + n) * 66 + kk];
      #pragma unroll
      for (int j = 0; j < 6; ++j) {
        v2f b;
        b.x = Bs[kk * 100 + j * 16 + n];
        b.y = Bs[(kk + 1) * 100 + j * 16 + n];
        acc[j] = WMMA_F32_16x16x4(a, b, acc[j]);
      }
    }
    __syncthreads();
  }

  #pragma unroll
  for (int j = 0; j < 6; ++j) {
    int col = colBase + j * 16 + n;
    float bias = (col < Y_) ? bD[col] : bV[col - Y_];
    #pragma unroll
    for (int v = 0; v < 8; ++v) {
      int row = rowBase + wrow + mh + v;
      float val = acc[j][v] + bias;
      if (col < Y_) {                                  // uniform per j
        float sp = (val > 20.0f) ? val : log1pf(expf(val));
        val = sp + 1e-3f;
      }
      DV[(size_t)row * DVW + col] = val;
    }
  }
}

// ---------------------------------------------------------------------------
// Kernel 3: per batch element — Woodbury + Newton–Schulz sqrt via WMMA.
// One wave per batch element; 2 waves per block; matrices LDS-resident.
// ---------------------------------------------------------------------------
constexpr int BUFS  = 34;         // padded row stride (bank-conflict-free A frags)
constexpr int BUFSZ = 32 * BUFS;  // 1088 floats per matrix buffer
constexpr int NS_ITERS = 30;

// D = alpha * (A @ Bm) + beta_diag * I     (all 32x32, LDS, single wave)
__device__ __forceinline__ void mm32(float* __restrict__ D,
                                     const float* __restrict__ A,
                                     const float* __restrict__ Bm,
                                     int lane, float alpha, float beta_diag) {
  const int n  = lane & 15;
  const int kh = (lane >> 4) << 1;
  const int mh = (lane >> 4) << 3;
  #pragma unroll
  for (int ti = 0; ti < 2; ++ti) {
    #pragma unroll
    for (int tj = 0; tj < 2; ++tj) {
      v8f acc = {};
      #pragma unroll
      for (int kt = 0; kt < 8; ++kt) {            // K = 32 in steps of 4
        int kk = kt * 4 + kh;
        v2f a = *(const v2f*)&A[(ti * 16 + n) * BUFS + kk];
        v2f b;
        b.x = Bm[kk * BUFS + tj * 16 + n];
        b.y = Bm[(kk + 1) * BUFS + tj * 16 + n];
        acc = WMMA_F32_16x16x4(a, b, acc);
      }
      #pragma unroll
      for (int v = 0; v < 8; ++v) {
        int row = ti * 16 + mh + v;
        int col = tj * 16 + n;
        float val = alpha * acc[v] + ((row == col) ? beta_diag : 0.0f);
        D[row * BUFS + col] = val;
      }
    }
  }
}

__global__ __launch_bounds__(64) void k_sqrt(const float* __restrict__ DV,
                                             float* __restrict__ out) {
  __shared__ __align__(16) float smem[2 * 4 * BUFSZ];   // 34.8 KB
  const int wave = threadIdx.x >> 5;
  const int lane = threadIdx.x & 31;
  const int b = blockIdx.x * 2 + wave;
  float* S  = &smem[wave * 4 * BUFSZ];
  float* Yb = S;                 // buf0: Y
  float* Zb = S + BUFSZ;         // buf1: Z (M scratch first)
  float* Vm = S + 2 * BUFSZ;     // buf2: V rows (stride 9), later T
  float* Wm = S + 3 * BUFSZ;     // buf3: W rows (stride 9), later U

  const float* dvb = DV + (size_t)b * DVW;
  float Dl    = dvb[lane];                 // lane = y (0..31)
  float invd2 = 1.0f / (Dl * Dl);

  float vr[8], wr[8];
  #pragma unroll
  for (int j = 0; j < K_; ++j) {
    vr[j] = dvb[Y_ + lane * K_ + j];       // V[b][lane][j]
    wr[j] = vr[j] * invd2;                 // W = A_inv V
    Vm[lane * 9 + j] = vr[j];
    Wm[lane * 9 + j] = wr[j];
  }
  __builtin_amdgcn_wave_barrier();

  // M = I + V^T W  (8x8): each lane computes 2 entries into Zb[0..63]
  #pragma unroll
  for (int t = 0; t < 2; ++t) {
    int e = lane + t * 32;
    int k = e >> 3, j = e & 7;
    float s = (k == j) ? 1.0f : 0.0f;
    #pragma unroll
    for (int y = 0; y < Y_; ++y) s += Vm[y * 9 + k] * Wm[y * 9 + j];
    Zb[e] = s;
  }
  __builtin_amdgcn_wave_barrier();

  // 8x8 Cholesky + inverse, computed redundantly in all lanes (fully unrolled)
  float M8[8][8];
  #pragma unroll
  for (int i = 0; i < 8; ++i)
    #pragma unroll
    for (int j = 0; j < 8; ++j) M8[i][j] = Zb[i * 8 + j];

  float L[8][8];
  #pragma unroll
  for (int i = 0; i < 8; ++i)
    #pragma unroll
    for (int j = 0; j < 8; ++j) L[i][j] = 0.0f;
  #pragma unroll
  for (int j = 0; j < 8; ++j) {
    float s = M8[j][j];
    #pragma unroll
    for (int p = 0; p < j; ++p) s -= L[j][p] * L[j][p];
    float dj = sqrtf(fmaxf(s, 1e-20f));
    L[j][j] = dj;
    float inv = 1.0f / dj;
    #pragma unroll
    for (int i = j + 1; i < 8; ++i) {
      float t = M8[i][j];
      #pragma unroll
      for (int p = 0; p < j; ++p) t -= L[i][p] * L[j][p];
      L[i][j] = t * inv;
    }
  }
  float Li[8][8];
  #pragma unroll
  for (int i = 0; i < 8; ++i)
    #pragma unroll
    for (int j = 0; j < 8; ++j) Li[i][j] = 0.0f;
  #pragma unroll
  for (int i = 0; i < 8; ++i) {
    float di = 1.0f / L[i][i];
    Li[i][i] = di;
    #pragma unroll
    for (int j = 0; j < i; ++j) {
      float s = 0.0f;
      #pragma unroll
      for (int p = j; p < i; ++p) s += L[i][p] * Li[p][j];
      Li[i][j] = -di * s;
    }
  }
  float Mi[8][8];                           // M_inv = Li^T Li
  #pragma unroll
  for (int i = 0; i < 8; ++i)
    #pragma unroll
    for (int j = 0; j < 8; ++j) {
      float s = 0.0f;
      #pragma unroll
      for (int p = 0; p < 8; ++p) s += Li[p][i] * Li[p][j];
      Mi[i][j] = s;
    }

  // u = w_row * M_inv  (per lane = per row y)
  float u[8];
  #pragma unroll
  for (int j = 0; j < 8; ++j) {
    float s = 0.0f;
    #pragma unroll
    for (int k = 0; k < 8; ++k) s += wr[k] * Mi[k][j];
    u[j] = s;
  }

  // Sigma_inv row y (lane) and Frobenius norm
  float ar[32];
  float fro = 0.0f;
  #pragma unroll
  for (int z = 0; z < 32; ++z) {
    float s = 0.0f;
    #pragma unroll
    for (int j = 0; j < 8; ++j) s += u[j] * Wm[z * 9 + j];
    float val = ((z == lane) ? invd2 : 0.0f) - s;
    ar[z] = val;
    fro += val * val;
  }
  #pragma unroll
  for (int off = 16; off > 0; off >>= 1) fro += __shfl_xor(fro, off, 32);
  float anorm = sqrtf(fro);
  float sc = 1.0f / anorm;

  // Y0 = A/||A||_F ; Z0 = I
  #pragma unroll
  for (int z = 0; z < 32; ++z) {
    Yb[lane * BUFS + z] = ar[z] * sc;
    Zb[lane * BUFS + z] = (z == lane) ? 1.0f : 0.0f;
  }
  __builtin_amdgcn_wave_barrier();

  // Coupled Newton–Schulz: T = 1.5I - 0.5*Z*Y ; Y <- Y*T ; Z <- T*Z
  float *pY = Yb, *pZ = Zb, *pT = Vm, *pU = Wm;
  for (int iter = 0; iter < NS_ITERS; ++iter) {
    mm32(pT, pZ, pY, lane, -0.5f, 1.5f);
    mm32(pU, pY, pT, lane, 1.0f, 0.0f);   // new Y
    mm32(pY, pT, pZ, lane, 1.0f, 0.0f);   // new Z (into old Y slot)
    float* nY = pU; float* nZ = pY; float* nT = pZ; float* nU = pT;
    pY = nY; pZ = nZ; pT = nT; pU = nU;
  }
  __builtin_amdgcn_wave_barrier();

  // Sigma_inv_sqrt = sqrt(||A||_F) * Y
  float sq = sqrtf(anorm);
  float* ob = out + (size_t)b * (Y_ * Y_);
  #pragma unroll
  for (int r = 0; r < 32; ++r)
    ob[r * 32 + lane] = sq * pY[r * BUFS + lane];
}

// ---------------------------------------------------------------------------
extern "C" void kernel_launch(void* const* d_in, const int* in_sizes, int n_in,
                              void* d_out, int out_size, void* d_ws, size_t ws_size,
                              hipStream_t stream) {
  (void)in_sizes; (void)n_in; (void)out_size; (void)ws_size;
  const float* x  = (const float*)d_in[0];
  const float* W1 = (const float*)d_in[1];
  const float* b1 = (const float*)d_in[2];
  const float* WD = (const float*)d_in[3];
  const float* bD = (const float*)d_in[4];
  const float* WV = (const float*)d_in[5];
  const float* bV = (const float*)d_in[6];
  float* out  = (float*)d_out;
  float* feat = out;                 // reuse d_out as scratch for feat [B,H]
  float* DV   = (float*)d_ws;        // [B,288] in workspace (18.9 MB)

  k_feat<<<dim3(B_ / 64, H_ / 64), 128, 0, stream>>>(x, W1, b1, feat);
  k_head<<<dim3(B_ / 64, 3), 128, 0, stream>>>(feat, WD, bD, WV, bV, DV);
  k_sqrt<<<dim3(B_ / 2), 64, 0, stream>>>(DV, out);
}